// TangetBundle_partition_of_unity_atlas_3813930959558
// MI455X (gfx1250) — compile-verified
//
#include <hip/hip_runtime.h>
#include <cmath>

typedef float v2f __attribute__((ext_vector_type(2)));
typedef float v8f __attribute__((ext_vector_type(8)));
typedef unsigned int v4u __attribute__((ext_vector_type(4)));
typedef int v4i __attribute__((ext_vector_type(4)));
typedef int v8i __attribute__((ext_vector_type(8)));

// ---------------------------------------------------------------------------
// D = A(16x4,f32) * B(4x16,f32) + C(16x16,f32)  -- V_WMMA_F32_16X16X4_F32
// A frag: lane L holds A[L%16][k0+2*(L/16)+{0,1}]
// B frag: lane L holds B[k0+2*(L/16)+{0,1}][L%16]
// C frag: vgpr r holds row (r + 8*(L/16)), col (L%16)
// ---------------------------------------------------------------------------
static __device__ __forceinline__ v8f wmma_f32_16x16x4(float a0, float a1,
                                                       float b0, float b1, v8f c) {
  v2f A; A[0] = a0; A[1] = a1;
  v2f B; B[0] = b0; B[1] = b1;
  return __builtin_amdgcn_wmma_f32_16x16x4_f32(false, A, false, B, (short)0, c,
                                               false, false);
}

#if __has_builtin(__builtin_amdgcn_tensor_load_to_lds)
#define USE_TDM 1
// ---------------------------------------------------------------------------
// Tensor Data Mover: 2-D tile global->LDS per CDNA5 ISA.
// Group0: count=1 | lds_addr | global_addr[56:0] | type=2.
// Group1: data_size=2 (4B), pad fields, tensor_dim0/1, tile_dim0/1, stride0.
// Groups 2/3 zero (<=2-D), cpol=0.  Tracked on TENSORcnt.
// This toolchain uses the 6-arg builtin form (extra int32x8 group, then cpol).
// ---------------------------------------------------------------------------
static __device__ __forceinline__ void tdm_load_2d(unsigned lds_off, const void* gsrc,
                                                   unsigned td0, unsigned td1,
                                                   unsigned stride0, unsigned pad_en,
                                                   unsigned pad_int, unsigned pad_amt) {
  unsigned long long ga = (unsigned long long)(size_t)gsrc;
  v4u g0;
  g0[0] = 1u;                                              // count=1 (valid user D#)
  g0[1] = lds_off;                                         // lds_addr (bytes)
  g0[2] = (unsigned)(ga & 0xffffffffu);                    // global_addr[31:0]
  g0[3] = (unsigned)((ga >> 32) & 0x01ffffffu) | 0x80000000u; // addr[56:32] | type=2
  v8i g1;
  g1[0] = (int)((2u << 16) | (pad_en << 20) | (pad_int << 22) | (pad_amt << 25));
  g1[1] = (int)((td0 & 0xffffu) << 16);                    // tensor_dim0[15:0]
  g1[2] = (int)(((td0 >> 16) & 0xffffu) | ((td1 & 0xffffu) << 16));
  g1[3] = (int)(((td1 >> 16) & 0xffffu) | ((td0 & 0xffffu) << 16)); // tile_dim0 = td0
  g1[4] = (int)(td1 & 0xffffu);                            // tile_dim1 = td1, tile_dim2 = 0
  g1[5] = (int)stride0;                                    // tensor_dim0_stride[31:0]
  g1[6] = 0;
  g1[7] = 0;
  v4i z4; z4[0] = 0; z4[1] = 0; z4[2] = 0; z4[3] = 0;
  v8i z8; z8[0] = 0; z8[1] = 0; z8[2] = 0; z8[3] = 0;
  z8[4] = 0; z8[5] = 0; z8[6] = 0; z8[7] = 0;
  __builtin_amdgcn_tensor_load_to_lds(g0, g1, z4, z4, z8, 0);
}
#else
#define USE_TDM 0
#endif

#define NTHREADS 64

__global__ __launch_bounds__(NTHREADS) void atlas_kernel(
    const float* __restrict__ y_in, const float* __restrict__ centers,
    const float* __restrict__ W_enc, const float* __restrict__ b_enc,
    const float* __restrict__ W_dec, const float* __restrict__ b_dec,
    const float* __restrict__ A_met, const float* __restrict__ W_met,
    const float* __restrict__ t_p, const int* __restrict__ ns_p,
    float* __restrict__ out) {
  // LDS (bank-conflict padded); total ~49.7 KB
  __shared__ float sWenc[16 * 68];     // W_enc[n][k], pitch 68 (4-bank shift/row)
  __shared__ float sBenc[16];
  __shared__ float scent[8];
  __shared__ float sWm[512];           // W_met[c][i][j]
  __shared__ float sBase[512];         // base_c = A_c A_c^T + I
  __shared__ float sBdec[512];         // b_dec[c][n]
  __shared__ float sZ[64 * 16];        // z per point
  __shared__ float sP[64 * 65];        // p[pt][i*8+c], pitch 65 (1-bank shift/row)
  __shared__ float sBuf[2][8 * 16 * 18]; // per-wave decode staging, pitch 18

  const int tid = threadIdx.x;
  const int lane = tid & 31;
  const int wv = tid >> 5;
  const int blockBase = blockIdx.x * NTHREADS;

  __builtin_prefetch(y_in + (size_t)(blockBase + tid) * 64, 0, 0); // global_prefetch_b8

  // ---- preload constants into LDS ----
#if USE_TDM
  if (wv == 0) {
    // W_enc: 16 rows x 64 DW, LDS-padded to pitch 68 (pad_interval 5 => every 64
    // DWORDs, pad_amount 3 => 4 DWORDs) -- TDM reproduces the bank-pad layout.
    tdm_load_2d((unsigned)(size_t)&sWenc[0], W_enc, 64, 16, 64, 1, 5, 3);
    tdm_load_2d((unsigned)(size_t)&sWm[0], W_met, 512, 1, 512, 0, 0, 0);
    tdm_load_2d((unsigned)(size_t)&sBdec[0], b_dec, 512, 1, 512, 0, 0, 0);
  }
#else
  for (int idx = tid; idx < 16 * 64; idx += NTHREADS)
    sWenc[(idx >> 6) * 68 + (idx & 63)] = W_enc[idx];
  for (int idx = tid; idx < 512; idx += NTHREADS) sWm[idx] = W_met[idx];
  for (int idx = tid; idx < 512; idx += NTHREADS) sBdec[idx] = b_dec[idx];
#endif
  if (tid < 16) sBenc[tid] = b_enc[tid];
  if (tid < 8) scent[tid] = centers[tid];
  for (int e = tid; e < 512; e += NTHREADS) {
    int c = e >> 6, rem = e & 63, i = rem >> 3, k = rem & 7;
    const float* Ai = A_met + c * 64 + i * 8;
    const float* Ak = A_met + c * 64 + k * 8;
    float s = (i == k) ? 1.0f : 0.0f;
#pragma unroll
    for (int j = 0; j < 8; ++j) s += Ai[j] * Ak[j];
    sBase[e] = s;
  }
#if __has_builtin(__builtin_amdgcn_s_wait_tensorcnt)
  __builtin_amdgcn_s_wait_tensorcnt(0); // TDM data landed before the barrier
#endif
  __syncthreads();

  // =======================================================================
  // Phase A: encoder Z = Y * W_enc^T + b_enc via WMMA f32 16x16x4 (K=64)
  // =======================================================================
  {
    const int m = lane & 15, hh = lane >> 4;
#pragma unroll
    for (int t16 = 0; t16 < 2; ++t16) {
      const int tileG = blockBase + wv * 32 + t16 * 16;
      const float* yrow = y_in + (size_t)(tileG + m) * 64;
      v8f cz = {0.f, 0.f, 0.f, 0.f, 0.f, 0.f, 0.f, 0.f};
#pragma unroll
      for (int q = 0; q < 16; ++q) {
        const int k = 4 * q + 2 * hh;
        cz = wmma_f32_16x16x4(yrow[k], yrow[k + 1],
                              sWenc[m * 68 + k], sWenc[m * 68 + k + 1], cz);
      }
#pragma unroll
      for (int r = 0; r < 8; ++r) {
        const int pt = r + 8 * hh;
        sZ[(wv * 32 + t16 * 16 + pt) * 16 + m] = cz[r] + sBenc[m];
      }
    }
  }
  __syncthreads();

  // =======================================================================
  // Phase B: per-thread RK4 geodesic integration (8-dim manifold)
  // dv = -0.5 * g^{-1} (2*P1 - P3), P1=(D_v g)v, P3=grad_x(v^T g v)
  // =======================================================================
  {
    float zc[16];
#pragma unroll
    for (int d = 0; d < 16; ++d) zc[d] = sZ[tid * 16 + d];
    const int ns = ns_p[0];
    const float dt = t_p[0] / (float)ns;

#pragma clang loop unroll(disable)
    for (int stp = 0; stp < ns; ++stp) {
      float acc[16], zt[16];
#pragma unroll
      for (int d = 0; d < 16; ++d) { acc[d] = 0.f; zt[d] = zc[d]; }

#pragma clang loop unroll(disable)
      for (int s = 0; s < 4; ++s) {
        float kk[16];
        { // ---- ODE(zt) -> kk ----
          float ssum = 0.f, sv = 0.f;
#pragma unroll
          for (int j = 0; j < 8; ++j) { ssum += zt[j]; sv += zt[8 + j]; }
          // softmax stats over charts
          float fmx = -1e30f;
#pragma clang loop unroll(disable)
          for (int c = 0; c < 8; ++c) { float d = ssum - scent[c]; fmx = fmaxf(fmx, -d * d); }
          float esum = 0.f;
#pragma clang loop unroll(disable)
          for (int c = 0; c < 8; ++c) { float d = ssum - scent[c]; esum += __expf(-d * d - fmx); }
          const float inv_es = 1.0f / esum;
          float fbar = 0.f;
#pragma clang loop unroll(disable)
          for (int c = 0; c < 8; ++c) {
            float d = ssum - scent[c];
            fbar += (__expf(-d * d - fmx) * inv_es) * (-2.0f * d);
          }

          float gL[36], P1[8], t3[8], S = 0.f;
#pragma unroll
          for (int i = 0; i < 36; ++i) gL[i] = 0.f;
#pragma unroll
          for (int i = 0; i < 8; ++i) { P1[i] = 0.f; t3[i] = 0.f; }

#pragma clang loop unroll(disable)
          for (int c = 0; c < 8; ++c) {
            const float d = ssum - scent[c];
            const float w = __expf(-d * d - fmx) * inv_es;
            const float wp = w * (-2.0f * d - fbar);
            const float* Wc = sWm + c * 64;
            const float* Bc = sBase + c * 64;
            float u[8], du[8], wtr[8];
#pragma unroll
            for (int j = 0; j < 8; ++j) wtr[j] = 0.f;
            float uv = 0.f, duv = 0.f;
#pragma unroll
            for (int i = 0; i < 8; ++i) {
              float rw[8];
              *reinterpret_cast<float4*>(&rw[0]) = *reinterpret_cast<const float4*>(Wc + i * 8);
              *reinterpret_cast<float4*>(&rw[4]) = *reinterpret_cast<const float4*>(Wc + i * 8 + 4);
              float wx = 0.f, wvv = 0.f;
#pragma unroll
              for (int j = 0; j < 8; ++j) { wx += rw[j] * zt[j]; wvv += rw[j] * zt[8 + j]; }
              const float ui = tanhf(wx);
              const float m2 = 1.0f - ui * ui;
              const float dui = m2 * wvv;
              const float ri = zt[8 + i] * m2;
              u[i] = ui; du[i] = dui;
              uv += ui * zt[8 + i]; duv += dui * zt[8 + i];
#pragma unroll
              for (int j = 0; j < 8; ++j) wtr[j] += rw[j] * ri; // W^T (v*(1-u^2))
            }
            float vbv = 0.f, bv[8];
#pragma unroll
            for (int i = 0; i < 8; ++i) {
              float rw[8];
              *reinterpret_cast<float4*>(&rw[0]) = *reinterpret_cast<const float4*>(Bc + i * 8);
              *reinterpret_cast<float4*>(&rw[4]) = *reinterpret_cast<const float4*>(Bc + i * 8 + 4);
              float bvi = 0.f;
#pragma unroll
              for (int k = 0; k < 8; ++k) bvi += rw[k] * zt[8 + k];
              bv[i] = bvi; vbv += zt[8 + i] * bvi;
              const float wui = w * u[i];
#pragma unroll
              for (int k = 0; k <= i; ++k)
                gL[i * (i + 1) / 2 + k] += w * rw[k] + wui * u[k]; // g (lower)
            }
            const float Qc = vbv + uv * uv;
            S += wp * Qc;
            const float dw = wp * sv;
            const float c2 = 2.0f * w * uv;
#pragma unroll
            for (int i = 0; i < 8; ++i) {
              const float Gv = bv[i] + u[i] * uv;
              P1[i] += dw * Gv + w * (du[i] * uv + u[i] * duv);
              t3[i] += c2 * wtr[i];
            }
          }
          float rhs[8];
#pragma unroll
          for (int i = 0; i < 8; ++i) rhs[i] = 2.0f * P1[i] - (S + t3[i]);
          // Cholesky of SPD g (packed lower), then solve g y = rhs
          float invd[8];
#pragma unroll
          for (int j = 0; j < 8; ++j) {
            float djj = gL[j * (j + 1) / 2 + j];
#pragma unroll
            for (int k = 0; k < j; ++k) { float l = gL[j * (j + 1) / 2 + k]; djj -= l * l; }
            const float ljj = sqrtf(djj);
            gL[j * (j + 1) / 2 + j] = ljj;
            invd[j] = 1.0f / ljj;
#pragma unroll
            for (int i = j + 1; i < 8; ++i) {
              float tv = gL[i * (i + 1) / 2 + j];
#pragma unroll
              for (int k = 0; k < j; ++k) tv -= gL[i * (i + 1) / 2 + k] * gL[j * (j + 1) / 2 + k];
              gL[i * (i + 1) / 2 + j] = tv * invd[j];
            }
          }
          float w1[8];
#pragma unroll
          for (int i = 0; i < 8; ++i) {
            float tv = rhs[i];
#pragma unroll
            for (int k = 0; k < i; ++k) tv -= gL[i * (i + 1) / 2 + k] * w1[k];
            w1[i] = tv * invd[i];
          }
          float w2[8];
#pragma unroll
          for (int i = 7; i >= 0; --i) {
            float tv = w1[i];
#pragma unroll
            for (int k = i + 1; k < 8; ++k) tv -= gL[k * (k + 1) / 2 + i] * w2[k];
            w2[i] = tv * invd[i];
          }
#pragma unroll
          for (int i = 0; i < 8; ++i) { kk[i] = zt[8 + i]; kk[8 + i] = -0.5f * w2[i]; }
        }
        const float wgt = (s == 0 || s == 3) ? 1.0f : 2.0f;
        const float a = (s < 2) ? 0.5f * dt : dt;
#pragma unroll
        for (int d = 0; d < 16; ++d) { acc[d] += wgt * kk[d]; zt[d] = zc[d] + a * kk[d]; }
      }
#pragma unroll
      for (int d = 0; d < 16; ++d) zc[d] += (dt * (1.0f / 6.0f)) * acc[d];
    }

    // write back final z; compute p[i][c] = softmax_c(-(x_i - center_c)^2)
#pragma unroll
    for (int d = 0; d < 16; ++d) sZ[tid * 16 + d] = zc[d];
#pragma clang loop unroll(disable)
    for (int i = 0; i < 8; ++i) {
      const float xi = zc[i];
      float fmx = -1e30f;
      for (int c = 0; c < 8; ++c) { float d = xi - scent[c]; fmx = fmaxf(fmx, -d * d); }
      float es = 0.f;
      for (int c = 0; c < 8; ++c) { float d = xi - scent[c]; es += __expf(-d * d - fmx); }
      const float inv = 1.0f / es;
      for (int c = 0; c < 8; ++c) {
        float d = xi - scent[c];
        sP[tid * 65 + i * 8 + c] = __expf(-d * d - fmx) * inv;
      }
    }
  }
  __syncthreads();

  // =======================================================================
  // Phase C: decoder. y_all[pt][c][n] = sum_d z[pt][d] * W_dec[c][n][d]
  // via WMMA (K=16), then out[pt][i][n] = sum_c p[i][c]*(y_all + b_dec)
  // =======================================================================
  {
    const int m = lane & 15, hh = lane >> 4;
#pragma unroll
    for (int t16 = 0; t16 < 2; ++t16) {
      const int ptL = wv * 32 + t16 * 16;
#pragma clang loop unroll(disable)
      for (int n0i = 0; n0i < 4; ++n0i) {
        const int n0 = n0i * 16;
#pragma clang loop unroll(disable)
        for (int c = 0; c < 8; ++c) {
          v8f cf = {0.f, 0.f, 0.f, 0.f, 0.f, 0.f, 0.f, 0.f};
          const float* wd = W_dec + (size_t)(c * 64 + n0 + m) * 16; // B[d][m]
          const float* zrow = sZ + (ptL + m) * 16;                  // A[m][d]
#pragma unroll
          for (int q = 0; q < 4; ++q) {
            const int dd = 4 * q + 2 * hh;
            cf = wmma_f32_16x16x4(zrow[dd], zrow[dd + 1], wd[dd], wd[dd + 1], cf);
          }
#pragma unroll
          for (int r = 0; r < 8; ++r) {
            const int pt = r + 8 * hh;
            sBuf[wv][(c * 16 + pt) * 18 + m] = cf[r];
          }
        }
        __syncthreads();
        // combine: lane = (pt = m, col-half = hh)
        float yb[8][8];
#pragma unroll
        for (int c = 0; c < 8; ++c)
#pragma unroll
          for (int cc = 0; cc < 8; ++cc)
            yb[c][cc] = sBuf[wv][(c * 16 + m) * 18 + hh * 8 + cc] +
                        sBdec[c * 64 + n0 + hh * 8 + cc];
        const int ptG = blockBase + ptL + m;
#pragma clang loop unroll(disable)
        for (int i = 0; i < 8; ++i) {
          float pr[8];
#pragma unroll
          for (int c = 0; c < 8; ++c) pr[c] = sP[(ptL + m) * 65 + i * 8 + c];
          float o[8];
#pragma unroll
          for (int cc = 0; cc < 8; ++cc) {
            float a = 0.f;
#pragma unroll
            for (int c = 0; c < 8; ++c) a += pr[c] * yb[c][cc];
            o[cc] = a;
          }
          float* op = out + ((size_t)ptG * 8 + i) * 64 + n0 + hh * 8;
          *reinterpret_cast<float4*>(op) = make_float4(o[0], o[1], o[2], o[3]);
          *reinterpret_cast<float4*>(op + 4) = make_float4(o[4], o[5], o[6], o[7]);
        }
        __syncthreads();
      }
    }
  }
}

extern "C" void kernel_launch(void* const* d_in, const int* in_sizes, int n_in,
                              void* d_out, int out_size, void* d_ws, size_t ws_size,
                              hipStream_t stream) {
  const float* y_in = (const float*)d_in[0];
  const float* centers = (const float*)d_in[1];
  const float* W_enc = (const float*)d_in[2];
  const float* b_enc = (const float*)d_in[3];
  const float* W_dec = (const float*)d_in[4];
  const float* b_dec = (const float*)d_in[5];
  const float* A_met = (const float*)d_in[6];
  const float* W_met = (const float*)d_in[7];
  const float* t_p = (const float*)d_in[8];
  const int* ns_p = (const int*)d_in[9];
  float* out = (float*)d_out;

  const int npts = in_sizes[0] / 64; // B = 16384
  dim3 grid(npts / NTHREADS), block(NTHREADS);
  hipLaunchKernelGGL(atlas_kernel, grid, block, 0, stream, y_in, centers, W_enc,
                     b_enc, W_dec, b_dec, A_met, W_met, t_p, ns_p, out);
}